// GCNLayer_25898652795458
// MI455X (gfx1250) — compile-verified
//
#include <hip/hip_runtime.h>

// ---------------------------------------------------------------------------
// GCN layer for MI455X (gfx1250, wave32, WMMA + async global->LDS).
//   out = relu( D^-1/2 A D^-1/2 (xW^T + b) + x )
// Pipeline:
//   1) dis = rsqrt(rowsum(adj)+1e-6)                       (128 MiB pass)
//   2) supportS = dis[m] * (x @ Wt + b)    via V_WMMA_F32_16X16X4_F32
//   3) out = relu(dis[n] * (adj @ supportS) + x)  same WMMA GEMM, fused epilogue
// norm_adj never materialized (saves 256 MiB HBM traffic). f32 WMMA keeps
// reference precision (K=2048 contraction cancels heavily; bf16 would not).
// K-tiles are staged with GLOBAL_LOAD_ASYNC_TO_LDS_B128 (ASYNCcnt), double
// buffered; copies overlap WMMA compute and are fenced by s_wait_asynccnt 0
// before each barrier.
// ---------------------------------------------------------------------------

#include <stdint.h>

typedef float v2f __attribute__((ext_vector_type(2)));
typedef float v8f __attribute__((ext_vector_type(8)));

#define BM 128
#define BN 64
#define KT 32
#define PADA 4   // As row stride 36: b64 frag loads conflict-free, 16B-aligned B128 dests
#define PADB 4   // Bs row stride 68: float4-aligned rows

// ---- gfx1250 async global->LDS copy (B128) --------------------------------
// Builtin parameter types (from hipcc diagnostic): global int4* src, then
// (symmetrically) LDS int4* dst, int offset, int cpol.
typedef int v4i_vs __attribute__((vector_size(16)));
typedef v4i_vs __attribute__((address_space(1)))* gv4i_p;  // global
typedef v4i_vs __attribute__((address_space(3)))* lv4i_p;  // LDS

__device__ __forceinline__ void async_copy_b128(const float* g, float* l) {
#if __has_builtin(__builtin_amdgcn_global_load_async_to_lds_b128)
  __builtin_amdgcn_global_load_async_to_lds_b128(
      (gv4i_p)(unsigned long long)g,
      (lv4i_p)(unsigned)(unsigned long long)l, 0, 0);
#else
  unsigned loff = (unsigned)(unsigned long long)l;  // flat LDS addr low 32 = LDS offset
  asm volatile("global_load_async_to_lds_b128 %0, %1, off"
               :: "v"(loff), "v"(g)
               : "memory");
#endif
}

__device__ __forceinline__ void wait_async0() {
#if __has_builtin(__builtin_amdgcn_s_wait_asynccnt)
  __builtin_amdgcn_s_wait_asynccnt(0);
#else
  asm volatile("s_wait_asynccnt 0" ::: "memory");
#endif
}

// ---------------- degree / D^{-1/2} -----------------------------------------
__global__ __launch_bounds__(256) void degree_kernel(const float* __restrict__ adj,
                                                     float* __restrict__ dis, int N) {
  __shared__ float red[256];
  const float4* p4 = (const float4*)(adj + (long long)blockIdx.x * N);
  float s = 0.0f;
  for (int i = threadIdx.x; i < (N >> 2); i += 256) {
    float4 a = p4[i];
    s += (a.x + a.y) + (a.z + a.w);
  }
  red[threadIdx.x] = s;
  __syncthreads();
  for (int off = 128; off > 0; off >>= 1) {
    if (threadIdx.x < off) red[threadIdx.x] += red[threadIdx.x + off];
    __syncthreads();
  }
  if (threadIdx.x == 0) dis[blockIdx.x] = 1.0f / sqrtf(red[0] + 1e-6f);
}

// ---------------- W[o][i] -> Wt[i][o] (so GEMM B operand is row-major) ------
__global__ __launch_bounds__(256) void transpose256(const float* __restrict__ W,
                                                    float* __restrict__ Wt) {
  __shared__ float tile[32][33];
  int tx = threadIdx.x & 31, ty = threadIdx.x >> 5;  // 32x8
  int ibase = blockIdx.x * 32, obase = blockIdx.y * 32;
  for (int j = 0; j < 32; j += 8)
    tile[ty + j][tx] = W[(obase + ty + j) * 256 + ibase + tx];
  __syncthreads();
  for (int j = 0; j < 32; j += 8)
    Wt[(ibase + ty + j) * 256 + obase + tx] = tile[tx][ty + j];
}

// ---------------- WMMA f32 GEMM with fused GCN epilogues --------------------
// mode 0: out[r][c] = (acc + aux[c]) * dis[r]            (support, pre-scaled)
// mode 1: out[r][c] = max(acc * dis[r] + aux[r*ldo+c],0) (adj GEMM + residual)
__global__ __launch_bounds__(256) void gcn_gemm_wmma(
    const float* __restrict__ A, long long strideA, int lda,
    const float* __restrict__ B, long long strideB, int ldb,
    int K, const float* __restrict__ dis, const float* __restrict__ aux,
    float* __restrict__ out, int ldo, int rowsPerBatch, int mode) {
  // As m-major (stride 36): A frag = one aligned ds_load_b64 into the WMMA
  // operand pair; Bs k-major (stride 68): async B128 rows land contiguously.
  __shared__ __align__(16) float As[2][BM][KT + PADA];
  __shared__ __align__(16) float Bs[2][KT][BN + PADB];

  const int tid = threadIdx.x;
  const int lane = tid & 31;
  const int wid = tid >> 5;
  const int waverow = (wid & 3) * 32;   // 4 wave-rows x 32
  const int wavecol = (wid >> 2) * 32;  // 2 wave-cols x 32
  const int z = blockIdx.z;
  const int mblk = blockIdx.y * BM;
  const int nblk = blockIdx.x * BN;

  const float* Ab = A + (long long)z * strideA + (long long)mblk * lda;
  const float* Bb = B + (long long)z * strideB + nblk;

  const v8f vzero = {0.f, 0.f, 0.f, 0.f, 0.f, 0.f, 0.f, 0.f};
  v8f acc[2][2];
#pragma unroll
  for (int r = 0; r < 2; ++r)
#pragma unroll
    for (int c = 0; c < 2; ++c) acc[r][c] = vzero;

  // per-thread staging coordinates (4 A-chunks + 2 B-chunks of 16B each)
  // A: idx = tid + i*256 -> m = idx/8, c4 = idx%8  (row m, floats 4*c4..+3)
  // B: idx = tid + i*256 -> kr = idx/16, c4 = idx%16

  // -- prologue: async-stage K-tile 0 into buffer 0 ----------------------
#pragma unroll
  for (int i = 0; i < 4; ++i) {
    int idx = tid + i * 256, m = idx >> 3, c4 = idx & 7;
    async_copy_b128(Ab + (long long)m * lda + (c4 << 2), &As[0][m][c4 << 2]);
  }
#pragma unroll
  for (int i = 0; i < 2; ++i) {
    int idx = tid + i * 256, kr = idx >> 4, c4 = idx & 15;
    async_copy_b128(Bb + (long long)kr * ldb + (c4 << 2), &Bs[0][kr][c4 << 2]);
  }
  wait_async0();
  __syncthreads();

  const int nK = K / KT;
  const int koff = (lane >= 16) ? 2 : 0;  // ISA A/B halves hold K+0 / K+2
  const int ml = lane & 15;

  for (int kt = 0; kt < nK; ++kt) {
    const int buf = kt & 1;
    if (kt + 1 < nK) {
      // async-prefetch next K-tile into buf^1; last readers of buf^1 were
      // separated by the previous barrier, so the copy engine may write now.
      const float* An = Ab + (kt + 1) * KT;
      const float* Bn = Bb + (long long)(kt + 1) * KT * ldb;
      const int nb = buf ^ 1;
#pragma unroll
      for (int i = 0; i < 4; ++i) {
        int idx = tid + i * 256, m = idx >> 3, c4 = idx & 7;
        async_copy_b128(An + (long long)m * lda + (c4 << 2), &As[nb][m][c4 << 2]);
      }
#pragma unroll
      for (int i = 0; i < 2; ++i) {
        int idx = tid + i * 256, kr = idx >> 4, c4 = idx & 15;
        async_copy_b128(Bn + (long long)kr * ldb + (c4 << 2), &Bs[nb][kr][c4 << 2]);
      }
    }

    // -- compute: 8 k-substeps x 4 WMMA (2x2 tile grid per wave) ---------
#pragma unroll
    for (int kk = 0; kk < KT; kk += 4) {
      v2f af[2], bf[2];
#pragma unroll
      for (int r = 0; r < 2; ++r) {
        int m = waverow + r * 16 + ml;
        af[r] = *(const v2f*)&As[buf][m][kk + koff];  // ds_load_b64
      }
#pragma unroll
      for (int c = 0; c < 2; ++c) {
        int o = wavecol + c * 16 + ml;
        bf[c][0] = Bs[buf][kk + koff + 0][o];
        bf[c][1] = Bs[buf][kk + koff + 1][o];
      }
#pragma unroll
      for (int r = 0; r < 2; ++r)
#pragma unroll
        for (int c = 0; c < 2; ++c)
          acc[r][c] = __builtin_amdgcn_wmma_f32_16x16x4_f32(
              false, af[r], false, bf[c], (short)0, acc[r][c], false, false);
    }

    wait_async0();     // my copies resident before I signal
    __syncthreads();   // everyone's copies resident before anyone reads
  }

  // -- fused epilogue ----------------------------------------------------
  // C/D layout: VGPR v = row v (lanes 0-15) / row v+8 (lanes 16-31), col lane%16
  const long long growbase = (long long)z * rowsPerBatch + mblk;
#pragma unroll
  for (int r = 0; r < 2; ++r) {
#pragma unroll
    for (int c = 0; c < 2; ++c) {
      const int mb = waverow + r * 16 + ((lane >= 16) ? 8 : 0);
      const int col = nblk + wavecol + c * 16 + ml;
#pragma unroll
      for (int v = 0; v < 8; ++v) {
        const long long grow = growbase + mb + v;
        float val = acc[r][c][v];
        if (mode == 0)
          val = (val + aux[col]) * dis[grow];
        else
          val = fmaxf(fmaf(val, dis[grow], aux[grow * (long long)ldo + col]), 0.0f);
        out[grow * (long long)ldo + col] = val;
      }
    }
  }
}

// ---------------------------------------------------------------------------
extern "C" void kernel_launch(void* const* d_in, const int* in_sizes, int n_in,
                              void* d_out, int out_size, void* d_ws, size_t ws_size,
                              hipStream_t stream) {
  const float* x    = (const float*)d_in[0];  // [8,2048,256]
  const float* adj  = (const float*)d_in[1];  // [8,2048,2048]
  const float* W    = (const float*)d_in[2];  // [256,256]
  const float* bias = (const float*)d_in[3];  // [256]
  float* out = (float*)d_out;
  float* ws = (float*)d_ws;

  const int Bn = 8, N = 2048, D = 256;
  const long long rows = (long long)Bn * N;  // 16384

  float* supportS = ws;             // rows*D floats (16 MiB)
  float* dis      = ws + rows * D;  // rows floats
  float* Wt       = dis + rows;     // D*D floats

  // 1) degrees -> D^{-1/2}
  degree_kernel<<<dim3((unsigned)rows), dim3(256), 0, stream>>>(adj, dis, N);
  // 1.5) W^T so the linear GEMM's B operand is row-major [k][o]
  transpose256<<<dim3(8, 8), dim3(256), 0, stream>>>(W, Wt);
  // 2) supportS = dis[m] * (x @ Wt + b)   (M=16384,K=256,N=256)
  gcn_gemm_wmma<<<dim3(D / BN, (unsigned)(rows / BM), 1), dim3(256), 0, stream>>>(
      x, 0, D, Wt, 0, D, D, dis, bias, supportS, D, (int)rows, 0);
  // 3) out = relu(dis[n] * (adj @ supportS) + x)  (per batch M=2048,K=2048,N=256)
  gcn_gemm_wmma<<<dim3(D / BN, N / BM, Bn), dim3(256), 0, stream>>>(
      adj, (long long)N * N, N, supportS, (long long)N * D, D, N, dis, x, out, D, N, 1);
}